// By_Event_30717606101071
// MI455X (gfx1250) — compile-verified
//
#include <hip/hip_runtime.h>
#include <cstdint>
#include <cstddef>

// -----------------------------------------------------------------------------
// Event-based F1 (recall/precision/f1) for [B=512, L=61440] signals.
// Bandwidth-bound: 251.7 MB streamed once -> ~10.8us floor at 23.3 TB/s.
// One block per batch row. CDNA5 async global->LDS DMA (ASYNCcnt) double-buffers
// the HBM stream against bit-packing; then bit-packed run analysis + 128x128
// IoU mutual matching entirely in LDS.
// -----------------------------------------------------------------------------

#define L_LEN   61440
#define NWORDS  1920          // L_LEN / 32
#define EMAX    128
#define RAWMAX  2048          // cap on raw (pre-filter) events stored per row
#define LEN_THR 128
#define IOU_THR 0.2f
#define THREADS 256

// --- async streaming geometry ---
#define CHUNK      2048                   // floats per chunk per array
#define NCHUNK     (L_LEN / CHUNK)        // 30
#define CWORDS     (CHUNK / 32)           // 64 packed words per chunk
#define WSTRIDE_F  36                     // 32 data floats + 4 pad floats (144B)
#define STAGE_F    (CWORDS * WSTRIDE_F)   // 2304 floats per staging buffer

#if defined(__HIP_DEVICE_COMPILE__) && __has_builtin(__builtin_amdgcn_global_load_async_to_lds_b128)
#define HAVE_ASYNC_LDS 1
#else
#define HAVE_ASYNC_LDS 0
#endif

struct Phase2 {
    int rs[RAWMAX];               // raw start positions
    int re[RAWMAX];               // raw end positions
    int sO[EMAX], eO[EMAX];       // filtered+truncated output events
    int sT[EMAX], eT[EMAX];       // truncated target events
    int cnt[THREADS + 1];         // prefix-sum scratch
    float maxR[EMAX];  int idxR[EMAX];
    float maxC[EMAX];  int idxC[EMAX];
    float maxR2[EMAX]; int idxR2[EMAX];
    float maxC2[EMAX]; int idxC2[EMAX];
    int mR[EMAX], mC[EMAX], rOne[EMAX], cOne[EMAX];
    int m2[EMAX];
    int nOut, nTgt;
};

struct Stage1 {
    float stO[2][STAGE_F];        // double-buffered padded staging (output)
    float stT[2][STAGE_F];        // double-buffered padded staging (target)
};

struct SMem {
    uint32_t wO[NWORDS];          // packed binarized output
    uint32_t wT[NWORDS];          // packed binarized target
    union {
        Stage1 s1;                // phase 1: async DMA staging
        Phase2 p2;                // phases 2-5: events + matching
    } u;
};

#if HAVE_ASYNC_LDS
// Builtin signature (from hipcc diagnostic): pointee type is
// 'int __attribute__((vector_size(16)))' with global (as1) src, LDS (as3) dst.
typedef int v4i_vs __attribute__((vector_size(16)));
typedef __attribute__((address_space(1))) v4i_vs glob_v4i;
typedef __attribute__((address_space(3))) v4i_vs lds_v4i;

__device__ __forceinline__ void wait_async_le4() {
#if __has_builtin(__builtin_amdgcn_s_wait_asynccnt)
    __builtin_amdgcn_s_wait_asynccnt(4);
#else
    asm volatile("s_wait_asynccnt 0x4" ::: "memory");
#endif
}
__device__ __forceinline__ void wait_async_le0() {
#if __has_builtin(__builtin_amdgcn_s_wait_asynccnt)
    __builtin_amdgcn_s_wait_asynccnt(0);
#else
    asm volatile("s_wait_asynccnt 0x0" ::: "memory");
#endif
}

// Issue DMA for one 2048-float chunk of both arrays into staging buffers.
// Each lane copies 16B per op; 4 async ops per thread -> ASYNCcnt += 4 per wave.
__device__ __forceinline__ void issue_chunk(const float* orow, const float* trow,
                                            int c, float* stO, float* stT, int tid) {
    const float* go = orow + c * CHUNK;
    const float* gt = trow + c * CHUNK;
#pragma unroll
    for (int r = 0; r < 2; ++r) {
        const int f    = tid + THREADS * r;          // float4 index 0..511
        const int lofs = f * 4 + (f >> 3) * 4;       // padded float offset
        __builtin_amdgcn_global_load_async_to_lds_b128(
            (glob_v4i*)(go + f * 4), (lds_v4i*)(stO + lofs), 0, 0);
        __builtin_amdgcn_global_load_async_to_lds_b128(
            (glob_v4i*)(gt + f * 4), (lds_v4i*)(stT + lofs), 0, 0);
    }
}
#endif  // HAVE_ASYNC_LDS

__device__ __forceinline__ uint32_t trans_mask(const uint32_t* words, int w, bool ends) {
    uint32_t word = words[w];
    uint32_t prev = (w > 0) ? (words[w - 1] >> 31) : 0u;
    uint32_t sh   = (word << 1) | prev;
    // starts: b[pos]=1 && b[pos-1]=0 ; ends: b[pos]=0 && b[pos-1]=1
    return ends ? (~word & sh) : (word & ~sh);
}

// Extracts sorted transition positions into dst[0..RAWMAX). Returns FULL count
// (may exceed RAWMAX / EMAX, matching the reference's untruncated n).
__device__ int extract_events(const uint32_t* words, bool ends, int* dst, int* cnt) {
    const int tid = threadIdx.x;
    int c = 0;
    if (tid < 240) {
        const int w0 = tid * 8;
        for (int w = w0; w < w0 + 8; ++w) c += __popc(trans_mask(words, w, ends));
        if (ends && tid == 239 && (words[NWORDS - 1] >> 31)) c++;   // end at pos L
    }
    cnt[tid] = c;
    __syncthreads();
    if (tid == 0) {
        int s = 0;
        for (int i = 0; i < THREADS; ++i) { int v = cnt[i]; cnt[i] = s; s += v; }
        cnt[THREADS] = s;
    }
    __syncthreads();
    if (tid < 240) {
        int off = cnt[tid];
        const int w0 = tid * 8;
        for (int w = w0; w < w0 + 8; ++w) {
            uint32_t m = trans_mask(words, w, ends);
            while (m) {
                int i = __ffs((int)m) - 1;
                if (off < RAWMAX) dst[off] = w * 32 + i;
                off++;
                m &= m - 1;
            }
        }
        if (ends && tid == 239 && (words[NWORDS - 1] >> 31)) {
            if (off < RAWMAX) dst[off] = L_LEN;
            off++;
        }
    }
    int total = cnt[THREADS];
    __syncthreads();
    return total;
}

__device__ __forceinline__ float iou_val(const Phase2& s, int i, int j) {
    int so = s.sO[i], eo = s.eO[i], st = s.sT[j], et = s.eT[j];
    int inter = min(eo, et) - max(so, st);
    if (inter < 0) inter = 0;
    int den = (eo - so) + (et - st) - inter;
    if (den < 1) den = 1;
    bool valid = (i < s.nOut) && (j < s.nTgt);
    return valid ? (float)inter / (float)den : -1.0f;
}

__device__ __forceinline__ float iou2_val(const Phase2& s, int i, int j) {
    bool om = (((j == s.idxR[i]) && s.rOne[i]) || ((i == s.idxC[j]) && s.cOne[j]))
              && !(s.mR[i] || s.mC[j]);
    return om ? iou_val(s, i, j) : -1.0f;
}

__device__ __forceinline__ uint32_t pack_word_ge(const float4* p) {
    uint32_t b = 0;
#pragma unroll
    for (int j = 0; j < 8; ++j) {
        float4 a = p[j];
        b |= ((a.x >= 0.5f) ? 1u : 0u) << (4 * j + 0);
        b |= ((a.y >= 0.5f) ? 1u : 0u) << (4 * j + 1);
        b |= ((a.z >= 0.5f) ? 1u : 0u) << (4 * j + 2);
        b |= ((a.w >= 0.5f) ? 1u : 0u) << (4 * j + 3);
    }
    return b;
}

__device__ __forceinline__ uint32_t pack_word_ne(const float4* p) {
    uint32_t b = 0;
#pragma unroll
    for (int j = 0; j < 8; ++j) {
        float4 a = p[j];
        b |= ((a.x != 0.0f) ? 1u : 0u) << (4 * j + 0);
        b |= ((a.y != 0.0f) ? 1u : 0u) << (4 * j + 1);
        b |= ((a.z != 0.0f) ? 1u : 0u) << (4 * j + 2);
        b |= ((a.w != 0.0f) ? 1u : 0u) << (4 * j + 3);
    }
    return b;
}

__global__ void init_acc_kernel(int* acc) {
    if (threadIdx.x < 3) acc[threadIdx.x] = 0;
}

__global__ __launch_bounds__(THREADS) void event_f1_kernel(
        const float* __restrict__ out, const float* __restrict__ tgt,
        int* __restrict__ acc) {
    __shared__ SMem s;
    const int tid = threadIdx.x;
    const int row = blockIdx.x;
    const float* orow = out + (size_t)row * L_LEN;
    const float* trow = tgt + (size_t)row * L_LEN;

    // ---- Phase 1: stream + binarize + bit-pack (32 samples -> 1 word) ------
#if HAVE_ASYNC_LDS
    {
        int buf = 0;
        issue_chunk(orow, trow, 0, s.u.s1.stO[0], s.u.s1.stT[0], tid);
        for (int c = 0; c < NCHUNK; ++c) {
            const bool more = (c + 1 < NCHUNK);
            if (more)
                issue_chunk(orow, trow, c + 1, s.u.s1.stO[buf ^ 1], s.u.s1.stT[buf ^ 1], tid);
            if (more) wait_async_le4(); else wait_async_le0();   // chunk c DMA done
            __syncthreads();                                     // all waves' DMA visible
            if (tid < CWORDS) {
                const float4* p = (const float4*)(s.u.s1.stO[buf] + tid * WSTRIDE_F);
                s.wO[c * CWORDS + tid] = pack_word_ge(p);
            } else if (tid >= 128 && tid < 128 + CWORDS) {
                const int t2 = tid - 128;
                const float4* q = (const float4*)(s.u.s1.stT[buf] + t2 * WSTRIDE_F);
                s.wT[c * CWORDS + t2] = pack_word_ne(q);
            }
            __syncthreads();                                     // buffer free for reuse
            buf ^= 1;
        }
    }
#else
#pragma unroll
    for (int k = 0; k < 8; ++k) {
        int w = tid + THREADS * k;
        if (w < NWORDS) {
            if (w + THREADS < NWORDS) {                     // gfx1250 prefetch path
                __builtin_prefetch(orow + (w + THREADS) * 32, 0, 1);
                __builtin_prefetch(trow + (w + THREADS) * 32, 0, 1);
            }
            s.wO[w] = pack_word_ge((const float4*)(orow + w * 32));
            s.wT[w] = pack_word_ne((const float4*)(trow + w * 32));
        }
    }
    __syncthreads();
#endif

    Phase2& p2 = s.u.p2;

    // ---- Phase 2: event extraction ----------------------------------------
    // Target events (no length filter): first EMAX kept, padded with [L,L].
    int nT = extract_events(s.wT, false, p2.rs, p2.cnt);
    if (tid < EMAX) p2.sT[tid] = (tid < nT) ? p2.rs[tid] : L_LEN;
    __syncthreads();
    (void)extract_events(s.wT, true, p2.rs, p2.cnt);
    if (tid < EMAX) p2.eT[tid] = (tid < nT) ? p2.rs[tid] : L_LEN;
    __syncthreads();

    // Output raw events, then drop runs shorter than LEN_THR, then truncate.
    int nOs = extract_events(s.wO, false, p2.rs, p2.cnt);
    (void)extract_events(s.wO, true, p2.re, p2.cnt);
    if (tid == 0) {
        int nraw = min(nOs, RAWMAX);
        int kept = 0;
        for (int i = 0; i < nraw; ++i) {
            int st = p2.rs[i], en = p2.re[i];
            if (en - st >= LEN_THR) {
                if (kept < EMAX) { p2.sO[kept] = st; p2.eO[kept] = en; }
                kept++;
            }
        }
        for (int k = (kept < EMAX ? kept : EMAX); k < EMAX; ++k) {
            p2.sO[k] = L_LEN; p2.eO[k] = L_LEN;
        }
        p2.nOut = kept;     // full (untruncated) count, as in reference
        p2.nTgt = nT;
    }
    __syncthreads();

    // ---- Phase 3: pass-1 best match ---------------------------------------
    if (tid < EMAX) {                       // row max/argmax (first occurrence)
        const int i = tid;
        float best = -2.0f; int bidx = 0;
        for (int j = 0; j < EMAX; ++j) {
            float v = iou_val(p2, i, j);
            if (v > best) { best = v; bidx = j; }
        }
        p2.maxR[i] = best; p2.idxR[i] = bidx;
    } else {                                // column max/argmax
        const int j = tid - EMAX;
        float best = -2.0f; int bidx = 0;
        for (int i = 0; i < EMAX; ++i) {
            float v = iou_val(p2, i, j);
            if (v > best) { best = v; bidx = i; }
        }
        p2.maxC[j] = best; p2.idxC[j] = bidx;
    }
    __syncthreads();

    if (tid < EMAX) {
        const int i = tid;
        bool mr = (p2.maxR[i] >= IOU_THR) && (p2.idxC[p2.idxR[i]] == i);
        bool mc = (p2.maxC[i] >= IOU_THR) && (p2.idxR[p2.idxC[i]] == i);
        p2.mR[i] = mr ? 1 : 0;
        p2.mC[i] = mc ? 1 : 0;
        p2.rOne[i] = (!mr && (p2.maxR[i] >= IOU_THR)) ? 1 : 0;
        p2.cOne[i] = (!mc && (p2.maxC[i] >= IOU_THR)) ? 1 : 0;
    }
    __syncthreads();

    // ---- Phase 4: pass-2 best match on ones_m-masked IoU ------------------
    if (tid < EMAX) {
        const int i = tid;
        float best = -2.0f; int bidx = 0;
        for (int j = 0; j < EMAX; ++j) {
            float v = iou2_val(p2, i, j);
            if (v > best) { best = v; bidx = j; }
        }
        p2.maxR2[i] = best; p2.idxR2[i] = bidx;
    } else {
        const int j = tid - EMAX;
        float best = -2.0f; int bidx = 0;
        for (int i = 0; i < EMAX; ++i) {
            float v = iou2_val(p2, i, j);
            if (v > best) { best = v; bidx = i; }
        }
        p2.maxC2[j] = best; p2.idxC2[j] = bidx;
    }
    __syncthreads();

    if (tid < EMAX) {
        const int i = tid;
        p2.m2[i] = ((p2.maxR2[i] >= IOU_THR) && (p2.idxC2[p2.idxR2[i]] == i)) ? 1 : 0;
    }
    __syncthreads();

    // ---- Phase 5: per-row TP/FN/FP ----------------------------------------
    if (tid == 0) {
        int tp = 0;
        for (int i = 0; i < EMAX; ++i) tp += p2.mR[i] + p2.m2[i];
        const int n_out = p2.nOut, n_tgt = p2.nTgt;
        const bool t_empty = (n_tgt == 0);
        const bool o_empty = (!t_empty) && (n_out == 0);
        int tp_b = (t_empty || o_empty) ? 0 : tp;
        int fn_b = t_empty ? n_out : (o_empty ? 0 : (n_tgt - tp));
        int fp_b = t_empty ? 0 : (o_empty ? n_tgt : (n_out - tp));
        atomicAdd(&acc[0], tp_b);
        atomicAdd(&acc[1], fn_b);
        atomicAdd(&acc[2], fp_b);
    }
}

__global__ void finalize_kernel(const int* __restrict__ acc, float* __restrict__ out) {
    if (threadIdx.x == 0 && blockIdx.x == 0) {
        float TP = (float)acc[0], FN = (float)acc[1], FP = (float)acc[2];
        float recall    = (TP + FN == 0.0f) ? 0.0f : TP / fmaxf(TP + FN, 1.0f);
        float precision = (TP + FP == 0.0f) ? 0.0f : TP / fmaxf(TP + FP, 1.0f);
        float f1 = (precision + recall == 0.0f)
                       ? 0.0f
                       : 2.0f * precision * recall / fmaxf(precision + recall, 1e-30f);
        out[0] = recall;
        out[1] = precision;
        out[2] = f1;
    }
}

extern "C" void kernel_launch(void* const* d_in, const int* in_sizes, int n_in,
                              void* d_out, int out_size, void* d_ws, size_t ws_size,
                              hipStream_t stream) {
    (void)n_in; (void)out_size; (void)ws_size;
    const float* output = (const float*)d_in[0];
    const float* target = (const float*)d_in[1];
    float* out = (float*)d_out;
    int* acc = (int*)d_ws;                       // 3 ints of scratch

    const int rows = in_sizes[0] / L_LEN;        // 512

    init_acc_kernel<<<1, 32, 0, stream>>>(acc);
    event_f1_kernel<<<rows, THREADS, 0, stream>>>(output, target, acc);
    finalize_kernel<<<1, 32, 0, stream>>>(acc, out);
}